// GraphAttentionLayer_57226144251999
// MI455X (gfx1250) — compile-verified
//
#include <hip/hip_runtime.h>
#include <hip/hip_bf16.h>
#include <math.h>

// ---------------------------------------------------------------------------
// GAT layer for MI455X (gfx1250, wave32).
//   B=2, N=50000, E=800000, IN_F=128, OUT_F=64 (derived from sizes at launch).
// Roofline: GEMM is 1.6 GFLOP / ~77MB -> bandwidth-trivial; edge scatter is
// 102M f32 atomics into a 25.6MB L2-resident output -> L2 atomic bound.
// fp32 WMMA (16x16x4) used for the dense GEMM to keep fp32 numerics.
// ---------------------------------------------------------------------------

typedef __attribute__((ext_vector_type(2))) float v2f;
typedef __attribute__((ext_vector_type(8))) float v8f;

#define ALPHA_LRELU 0.2f
#define EPS_NORM    1e-6f

// ---------------- K0: zero d_out and norm ----------------
__global__ __launch_bounds__(256)
void k_zero(float* __restrict__ out, int nOut, float* __restrict__ norm, int nNorm) {
    int i = blockIdx.x * blockDim.x + threadIdx.x;
    if (i < nOut)  out[i]  = 0.0f;
    if (i < nNorm) norm[i] = 0.0f;
}

// ---------------- K1: Wh = h @ Wfc^T via V_WMMA_F32_16X16X4_F32 ------------
// One wave -> 16 rows x 64 cols.  K=128 in steps of 4 (32 chained WMMAs x4 tiles).
// A 16x4 frag: lane l holds rows M=l&15, K = kk + 2*(l>>4) .. +1  -> v2f
// B 4x16 frag: lane l holds col N=l&15, same K split              -> v2f
// C 16x16:     VGPR j -> M = j + 8*(l>>4), N = l&15
__global__ __launch_bounds__(256)
void k_gemm_wmma(const float* __restrict__ h, const float* __restrict__ Wfc,
                 float* __restrict__ Wh, int numRowTiles) {
    const int lane = threadIdx.x & 31;
    const int wave = (blockIdx.x * blockDim.x + threadIdx.x) >> 5;
    if (wave >= numRowTiles) return;          // wave-uniform: EXEC stays all-ones

    const int m0   = wave << 4;
    const int rowc = lane & 15;               // A: row index;  B: col index
    const int koff = (lane >> 4) << 1;        // 0 or 2 (half-wave K split)

    const float* aptr  = h   + (size_t)(m0 + rowc) * 128 + koff;
    const float* bptr0 = Wfc + (size_t)(rowc +  0) * 128 + koff;   // B[k][n]=Wfc[n][k]
    const float* bptr1 = Wfc + (size_t)(rowc + 16) * 128 + koff;
    const float* bptr2 = Wfc + (size_t)(rowc + 32) * 128 + koff;
    const float* bptr3 = Wfc + (size_t)(rowc + 48) * 128 + koff;

    v8f c0 = {}, c1 = {}, c2 = {}, c3 = {};
#pragma unroll
    for (int kk = 0; kk < 128; kk += 4) {
        v2f a  = *(const v2f*)(aptr  + kk);
        v2f b0 = *(const v2f*)(bptr0 + kk);
        v2f b1 = *(const v2f*)(bptr1 + kk);
        v2f b2 = *(const v2f*)(bptr2 + kk);
        v2f b3 = *(const v2f*)(bptr3 + kk);
        c0 = __builtin_amdgcn_wmma_f32_16x16x4_f32(false, a, false, b0, (short)0, c0, false, false);
        c1 = __builtin_amdgcn_wmma_f32_16x16x4_f32(false, a, false, b1, (short)0, c1, false, false);
        c2 = __builtin_amdgcn_wmma_f32_16x16x4_f32(false, a, false, b2, (short)0, c2, false, false);
        c3 = __builtin_amdgcn_wmma_f32_16x16x4_f32(false, a, false, b3, (short)0, c3, false, false);
    }

    const int nLo = lane & 15;
    const int mHi = (lane >> 4) << 3;         // 0 or 8
#pragma unroll
    for (int j = 0; j < 8; ++j) {
        float* r = Wh + (size_t)(m0 + mHi + j) * 64 + nLo;
        r[ 0] = c0[j];
        r[16] = c1[j];
        r[32] = c2[j];
        r[48] = c3[j];
    }
}

// ---------------- K2: per-node attention scalars ---------------------------
// alpha_src[i] = Wh[i,:]·a_src ; alpha_dst[i] = Wh[i,:]·a_dst   (i over B*N)
__global__ __launch_bounds__(256)
void k_alpha(const float* __restrict__ Wh, const float* __restrict__ Wattn,
             float* __restrict__ asrc, float* __restrict__ adst, int BN) {
    int i = blockIdx.x * blockDim.x + threadIdx.x;
    if (i >= BN) return;
    const float* row = Wh + (size_t)i * 64;
    float s = 0.0f, d = 0.0f;
#pragma unroll
    for (int f = 0; f < 64; f += 4) {
        float4 w = *(const float4*)(row + f);
        float4 a = *(const float4*)(Wattn + f);
        float4 b = *(const float4*)(Wattn + 64 + f);
        s += w.x * a.x + w.y * a.y + w.z * a.z + w.w * a.w;
        d += w.x * b.x + w.y * b.y + w.z * b.z + w.w * b.w;
    }
    asrc[i] = s;
    adst[i] = d;
}

// ---------------- K3: edge logits + softmax denominator --------------------
__global__ __launch_bounds__(256)
void k_edge_logits(const long long* __restrict__ ei,
                   const float* __restrict__ asrc, const float* __restrict__ adst,
                   float* __restrict__ eexp, float* __restrict__ norm,
                   int E, int N, int total) {
    int i = blockIdx.x * blockDim.x + threadIdx.x;    // over B*E
    if (i >= total) return;
    int b = i / E;
    int e = i - b * E;
    int s = (int)ei[e];
    int d = (int)ei[E + e];
    float v = asrc[b * N + s] + adst[b * N + d];
    v = (v > 0.0f) ? v : ALPHA_LRELU * v;             // leaky_relu
    float ee = expf(v);
    eexp[i] = ee;
    atomicAdd(norm + b * N + d, ee);
}

// ---------------- K4: wave-per-edge weighted scatter -----------------------
// 32 lanes x float2 = full 64-float row; coalesced 256B gather of Wh[src],
// 64 lane-parallel no-return global_atomic_add_f32 into out[dst].
__global__ __launch_bounds__(256)
void k_scatter(const long long* __restrict__ ei, const float* __restrict__ Wh,
               const float* __restrict__ eexp, const float* __restrict__ norm,
               float* __restrict__ out, int E, int N, int total) {
    const int lane  = threadIdx.x & 31;
    const int wave  = (blockIdx.x * blockDim.x + threadIdx.x) >> 5;
    const int waves = (gridDim.x * blockDim.x) >> 5;
    for (int i = wave; i < total; i += waves) {
        int b = i / E;
        int e = i - b * E;
        int s = (int)ei[e];
        int d = (int)ei[E + e];
        float attn = eexp[i] / (norm[b * N + d] + EPS_NORM);
        const float2 w = *(const float2*)(Wh + ((size_t)(b * N + s)) * 64 + 2 * lane);
        float* o = out + ((size_t)(b * N + d)) * 64 + 2 * lane;
        atomicAdd(o,     w.x * attn);
        atomicAdd(o + 1, w.y * attn);
    }
}

// ---------------- K5: ELU epilogue -----------------------------------------
__global__ __launch_bounds__(256)
void k_elu(float* __restrict__ out, int n) {
    int i = blockIdx.x * blockDim.x + threadIdx.x;
    if (i >= n) return;
    float x = out[i];
    out[i] = (x > 0.0f) ? x : expm1f(x);
}

// ---------------------------------------------------------------------------
extern "C" void kernel_launch(void* const* d_in, const int* in_sizes, int n_in,
                              void* d_out, int out_size, void* d_ws, size_t ws_size,
                              hipStream_t stream) {
    const float*     h     = (const float*)d_in[0];      // [B,N,128] f32
    const long long* ei    = (const long long*)d_in[1];  // [2,E] i64
    const float*     Wfc   = (const float*)d_in[2];      // [64,128] f32
    const float*     Wattn = (const float*)d_in[3];      // [1,128] f32
    float*           out   = (float*)d_out;              // [B,N,64] f32

    const int B  = 2;                       // per reference setup
    const int E  = in_sizes[1] / 2;
    const int BN = out_size / 64;           // B*N
    const int N  = BN / B;
    const int totalBE = B * E;

    // Workspace layout (floats)
    float* Wh   = (float*)d_ws;             // BN*64
    float* asrc = Wh   + (size_t)BN * 64;   // BN
    float* adst = asrc + BN;                // BN
    float* norm = adst + BN;                // BN
    float* eexp = norm + BN;                // B*E
    (void)ws_size; (void)n_in;

    // K0: zero accumulators (harness poisons once; graph replays need re-zero)
    {
        int n = out_size > BN ? out_size : BN;
        k_zero<<<(n + 255) / 256, 256, 0, stream>>>(out, out_size, norm, BN);
    }
    // K1: fp32 WMMA GEMM, one wave per 16-row tile (BN is a multiple of 16)
    {
        int tiles  = BN / 16;
        int blocks = (tiles + 7) / 8;       // 8 waves per 256-thread block
        k_gemm_wmma<<<blocks, 256, 0, stream>>>(h, Wfc, Wh, tiles);
    }
    // K2: per-node alpha scalars
    k_alpha<<<(BN + 255) / 256, 256, 0, stream>>>(Wh, Wattn, asrc, adst, BN);
    // K3: edge logits + atomic softmax denominator
    k_edge_logits<<<(totalBE + 255) / 256, 256, 0, stream>>>(ei, asrc, adst, eexp, norm,
                                                             E, N, totalBE);
    // K4: wave-per-edge weighted scatter (grid-stride over waves)
    k_scatter<<<8192, 256, 0, stream>>>(ei, Wh, eexp, norm, out, E, N, totalBE);
    // K5: ELU
    k_elu<<<(out_size + 255) / 256, 256, 0, stream>>>(out, out_size);
}